// NCCVxmLoss_19275813224854
// MI455X (gfx1250) — compile-verified
//
#include <hip/hip_runtime.h>

typedef __attribute__((ext_vector_type(2))) float v2f;
typedef __attribute__((ext_vector_type(8))) float v8f;

// Volume geometry (from reference: shape (2,1,160,192,160))
constexpr int VB = 2;
constexpr int VD = 160;   // z
constexpr int VH = 192;   // y
constexpr int VW = 160;   // x (contiguous)
constexpr long NVOX   = (long)VB * VD * VH * VW;   // 9,830,400
constexpr int  NLINES = VB * VD * VH;              // 61,440 x-lines
constexpr int  NTILES = VW / 16;                   // 10
constexpr float WINSZ = 729.0f;                    // 9^3

// ---------------------------------------------------------------------------
// Pass 1: compute {I, J, I*I, J*J, I*J} and 9-wide box sum along x via WMMA.
// The x box filter is a banded 160x160 matmul; each 16x16 output tile needs
// three constant 16x16 band blocks (left/center/right). Each wave handles 16
// lines (N dim) and sweeps the 10 x-tiles (M dim), K=16 as 4 chunks of K=4.
// ---------------------------------------------------------------------------
__global__ __launch_bounds__(256)
void ncc_pass_x(const float* __restrict__ predict,   // J
                const float* __restrict__ target,    // I
                float* __restrict__ ws)
{
    const int lane  = threadIdx.x & 31;
    const int wid   = threadIdx.x >> 5;
    const int group = blockIdx.x * 8 + wid;          // 0..3839 (16-line groups)
    const int m     = lane & 15;                     // A-row / N index
    const int khalf = lane >> 4;                     // 0: K=0,1  1: K=2,3

    // Constant band-matrix blocks in the documented 16x4-f32 A layout.
    v2f Al[4], Ac[4], Ar[4];
    #pragma unroll
    for (int kc = 0; kc < 4; ++kc) {
        #pragma unroll
        for (int r = 0; r < 2; ++r) {
            int k = kc * 4 + khalf * 2 + r;          // global K within 16
            int d = m - k; if (d < 0) d = -d;
            Ac[kc][r] = (d <= 4)        ? 1.0f : 0.0f;   // |m-k| <= 4
            Al[kc][r] = (k >= m + 12)   ? 1.0f : 0.0f;   // tile t-1 overlap
            Ar[kc][r] = (k <= m - 12)   ? 1.0f : 0.0f;   // tile t+1 overlap
        }
    }

    const long lin  = (long)group * 16 + m;          // this lane's line
    const long base = lin * VW;
    const float* Ip = target  + base;
    const float* Jp = predict + base;

    for (int t = 0; t < NTILES; ++t) {
        // Load the 3-tile x-window of I and J in B-operand (4x16) layout:
        // lanes 0-15 hold K=0,1 ; lanes 16-31 hold K=2,3 ; N = lane&15.
        v2f Ibuf[3][4], Jbuf[3][4];
        #pragma unroll
        for (int nb = 0; nb < 3; ++nb) {
            const int tt = t + nb - 1;
            const bool ok = (tt >= 0) && (tt < NTILES);   // zero (SAME) pad
            #pragma unroll
            for (int kc = 0; kc < 4; ++kc) {
                #pragma unroll
                for (int r = 0; r < 2; ++r) {
                    const int x = tt * 16 + kc * 4 + khalf * 2 + r;
                    Ibuf[nb][kc][r] = ok ? Ip[x] : 0.0f;
                    Jbuf[nb][kc][r] = ok ? Jp[x] : 0.0f;
                }
            }
        }

        #pragma unroll
        for (int c = 0; c < 5; ++c) {
            v8f acc = {};
            #pragma unroll
            for (int nb = 0; nb < 3; ++nb) {
                #pragma unroll
                for (int kc = 0; kc < 4; ++kc) {
                    const v2f iv = Ibuf[nb][kc];
                    const v2f jv = Jbuf[nb][kc];
                    v2f bop;
                    if      (c == 0) bop = iv;
                    else if (c == 1) bop = jv;
                    else if (c == 2) bop = iv * iv;
                    else if (c == 3) bop = jv * jv;
                    else             bop = iv * jv;
                    const v2f aop = (nb == 0) ? Al[kc] : ((nb == 1) ? Ac[kc] : Ar[kc]);
                    // D = A x B + C : banded box-sum matmul (EXEC all-1s here)
                    acc = __builtin_amdgcn_wmma_f32_16x16x4_f32(
                        false, aop, false, bop, (short)0, acc, false, false);
                }
            }
            // D layout: lanes 0-15 reg r -> M=r ; lanes 16-31 reg r -> M=8+r
            float* outp = ws + (long)c * NVOX + base + t * 16 + khalf * 8;
            #pragma unroll
            for (int r = 0; r < 8; ++r)
                outp[r] = acc[r];
        }
    }
}

// ---------------------------------------------------------------------------
// Pass 2: in-place rolling 9-sum along y for all 5 channels.
// Ring buffer keeps the raw inputs so the in-place update is hazard-free
// (out[y] is written only after x[y+4] was read; x[y-5] comes from the ring).
// ---------------------------------------------------------------------------
__global__ __launch_bounds__(256)
void ncc_pass_y(float* __restrict__ ws)
{
    const int tid = blockIdx.x * 256 + threadIdx.x;      // 0..255999
    const int x = tid % VW;
    const int z = (tid / VW) % VD;
    const int b = (tid / (VW * VD)) % VB;
    const int c =  tid / (VW * VD * VB);
    float* p = ws + (long)c * NVOX + (((long)b * VD + z) * VH) * VW + x;

    float ring[9];
    #pragma unroll
    for (int i = 0; i < 9; ++i) ring[i] = 0.0f;
    float s = 0.0f;
    #pragma unroll
    for (int pos = 0; pos < 4; ++pos) {                  // preload x[0..3]
        const float v = p[(long)pos * VW];
        s += v; ring[pos] = v;
    }
    for (int y = 0; y < VH; ++y) {
        const int pos = y + 4;
        const int idx = pos % 9;
        const float old = ring[idx];                     // == x[y-5] (or 0)
        const float v = (pos < VH) ? p[(long)pos * VW] : 0.0f;
        s += v - old;
        ring[idx] = v;
        p[(long)y * VW] = s;
    }
}

// ---------------------------------------------------------------------------
// Pass 3: rolling 9-sum along z (all 5 channels fused) + cc formula +
// deterministic block reduction (shfl + LDS, plain per-block store).
// ---------------------------------------------------------------------------
__global__ __launch_bounds__(256)
void ncc_pass_z_reduce(const float* __restrict__ ws,
                       float* __restrict__ blockSums)
{
    const int tid = blockIdx.x * 256 + threadIdx.x;      // 0..61439
    const int x = tid % VW;
    const int y = (tid / VW) % VH;
    const int b =  tid / (VW * VH);
    const long zstride = (long)VH * VW;
    const float* p0 = ws + ((long)b * VD * VH + y) * VW + x;

    float ring[5][9];
    float s[5];
    #pragma unroll
    for (int c = 0; c < 5; ++c) {
        s[c] = 0.0f;
        #pragma unroll
        for (int i = 0; i < 9; ++i) ring[c][i] = 0.0f;
    }
    #pragma unroll
    for (int pos = 0; pos < 4; ++pos) {
        #pragma unroll
        for (int c = 0; c < 5; ++c) {
            const float v = p0[(long)c * NVOX + (long)pos * zstride];
            s[c] += v; ring[c][pos] = v;
        }
    }

    float local = 0.0f;
    for (int z = 0; z < VD; ++z) {
        const int pos = z + 4;
        const int idx = pos % 9;
        #pragma unroll
        for (int c = 0; c < 5; ++c) {
            const float old = ring[c][idx];
            const float v = (pos < VD) ? p0[(long)c * NVOX + (long)pos * zstride] : 0.0f;
            s[c] += v - old;
            ring[c][idx] = v;
        }
        const float Is = s[0], Js = s[1], I2 = s[2], J2 = s[3], IJ = s[4];
        const float uI = Is / WINSZ;
        const float uJ = Js / WINSZ;
        const float cross = IJ - uJ * Is - uI * Js + uI * uJ * WINSZ;
        const float Ivar  = I2 - 2.0f * uI * Is + uI * uI * WINSZ;
        const float Jvar  = J2 - 2.0f * uJ * Js + uJ * uJ * WINSZ;
        local += cross * cross / (Ivar * Jvar + 1e-5f);
    }

    // wave32 reduction, then LDS across the 8 waves, fixed order -> deterministic
    #pragma unroll
    for (int off = 16; off > 0; off >>= 1)
        local += __shfl_down(local, off, 32);
    __shared__ float wsum[8];
    const int lane = threadIdx.x & 31;
    const int wid  = threadIdx.x >> 5;
    if (lane == 0) wsum[wid] = local;
    __syncthreads();
    if (threadIdx.x == 0) {
        float bs = 0.0f;
        #pragma unroll
        for (int w = 0; w < 8; ++w) bs += wsum[w];
        blockSums[blockIdx.x] = bs;
    }
}

__global__ void ncc_finalize(const float* __restrict__ blockSums,
                             float* __restrict__ out)
{
    if (threadIdx.x == 0 && blockIdx.x == 0) {
        float sum = 0.0f;
        for (int i = 0; i < 240; ++i) sum += blockSums[i];
        out[0] = 1.0f - sum / (float)NVOX;               // loss * LOSS_WEIGHT(=1)
    }
}

extern "C" void kernel_launch(void* const* d_in, const int* in_sizes, int n_in,
                              void* d_out, int out_size, void* d_ws, size_t ws_size,
                              hipStream_t stream) {
    const float* predict = (const float*)d_in[0];   // J
    const float* target  = (const float*)d_in[1];   // I
    float* ws        = (float*)d_ws;                // 5 channel volumes
    float* blockSums = ws + (size_t)5 * NVOX;       // +240 floats

    ncc_pass_x<<<NLINES / 16 / 8, 256, 0, stream>>>(predict, target, ws);   // 480 blocks
    ncc_pass_y<<<(5 * VB * VD * VW) / 256, 256, 0, stream>>>(ws);           // 1000 blocks
    ncc_pass_z_reduce<<<(VB * VH * VW) / 256, 256, 0, stream>>>(ws, blockSums); // 240 blocks
    ncc_finalize<<<1, 32, 0, stream>>>(blockSums, (float*)d_out);
}